// LeagueGNN_52862457479890
// MI455X (gfx1250) — compile-verified
//
#include <hip/hip_runtime.h>
#include <hip/hip_bf16.h>

// ---------------------------------------------------------------------------
// LeagueGNN forward on gfx1250 (MI455X), wave32.
// Dense node transforms use V_WMMA_F32_16X16X4_F32 (exact fp32), with A/B
// fragments staged through LDS (zero-padded) so the WMMA inner loop is
// branch-free: 2x ds_load_b64 + v_wmma per K-step.
// Edge aggregation uses float4 gathers + global f32 atomics (L2-resident).
// ---------------------------------------------------------------------------

typedef float v2f __attribute__((ext_vector_type(2)));
typedef float v8f __attribute__((ext_vector_type(8)));

// ---------------- WMMA GEMM: H[M,N] = X[M,K] @ W[K,N] (fp32) ----------------
// One 16-row M-tile per wave (4 waves/block); loops over 16-col N-tiles.
// Assumes M % 16 == 0 (true here: 50000 = 3125 * 16).
template <int K, int N>
__global__ __launch_bounds__(128)
void k_gemm_f32(const float* __restrict__ X, const float* __restrict__ W,
                float* __restrict__ H, int M) {
  constexpr int KP = (K + 3) & ~3;          // K padded to multiple of 4
  constexpr int WAVES = 4;
  __shared__ float sW[(KP / 2) * N * 2];    // packed K-pairs: [(k/2)*N + c] -> {W[k][c], W[k+1][c]}
  __shared__ float sA[WAVES * 16 * KP];     // per-wave 16xKP A tiles, zero-padded

  const int lane = threadIdx.x & 31;
  const int wave = threadIdx.x >> 5;
  const int mt   = blockIdx.x * WAVES + wave;
  const bool valid = (mt * 16 < M);         // wave-uniform

  // cooperative, zero-padded weight load (packed K-pairs for b64 B-fragment loads)
  for (int idx = threadIdx.x; idx < (KP / 2) * N; idx += 128) {
    const int k2 = idx / N, c = idx % N;
    const int k  = 2 * k2;
    float2 p;
    p.x = (k     < K) ? W[(size_t)k * N + c]       : 0.f;
    p.y = (k + 1 < K) ? W[(size_t)(k + 1) * N + c] : 0.f;
    *(float2*)(sW + 2 * idx) = p;
  }
  // per-wave A tile, zero-padded to KP
  if (valid) {
    float* a = sA + wave * 16 * KP;
    for (int idx = lane; idx < 16 * KP; idx += 32) {
      const int r = idx / KP, c = idx % KP;
      a[idx] = (c < K) ? X[(size_t)(mt * 16 + r) * K + c] : 0.f;
    }
  }
  __syncthreads();
  if (!valid) return;

  const int half = lane >> 4;               // 0: lanes 0-15, 1: lanes 16-31
  const int l16  = lane & 15;
  const float* myA = sA + wave * 16 * KP + l16 * KP;   // this lane's A row

  for (int nt = 0; nt < N / 16; ++nt) {
    const int col = nt * 16 + l16;
    v8f acc = {0.f, 0.f, 0.f, 0.f, 0.f, 0.f, 0.f, 0.f};
#pragma unroll
    for (int k0 = 0; k0 < KP; k0 += 4) {
      const int ka = k0 + 2 * half;                       // even
      const v2f a = *(const v2f*)(myA + ka);              // {A[r][ka], A[r][ka+1]}
      const v2f b = *(const v2f*)(sW + ka * N + 2 * col); // {W[ka][col], W[ka+1][col]}
      acc = __builtin_amdgcn_wmma_f32_16x16x4_f32(
          /*neg_a=*/false, a, /*neg_b=*/false, b,
          /*c_mod=*/(short)0, acc, /*reuse_a=*/false, /*reuse_b=*/false);
    }
    const int orow = mt * 16 + 8 * half;    // D rows: v + 8*half
#pragma unroll
    for (int v = 0; v < 8; ++v)
      H[(size_t)(orow + v) * N + col] = acc[v];
  }
}

// ---------------- degree / normalization ----------------
__global__ void k_deg(const int* __restrict__ dst, float* __restrict__ deg, int E) {
  int t = blockIdx.x * blockDim.x + threadIdx.x;
  if (t < E) atomicAdd(&deg[dst[t]], 1.0f);
}
__global__ void k_dinv(float* __restrict__ deg, int n) {
  int t = blockIdx.x * blockDim.x + threadIdx.x;
  if (t < n) deg[t] = rsqrtf(deg[t] + 1.0f);   // +1 for self-loop; always > 0
}

__global__ void k_fill(float* __restrict__ p, float v, long long total) {
  long long t = (long long)blockIdx.x * blockDim.x + threadIdx.x;
  if (t < total) p[t] = v;
}

// ---------------- GCN aggregation: out[dst] += hT[src] * dinv[s]*dinv[d] ----
template <int C>
__global__ void k_gcn_agg(const int* __restrict__ ei, const float* __restrict__ dinv,
                          const float* __restrict__ hT, float* __restrict__ out,
                          int E, int n) {
  constexpr int G = C / 4;
  long long t = (long long)blockIdx.x * blockDim.x + threadIdx.x;
  long long total = (long long)(E + n) * G;
  if (t >= total) return;
  int e = (int)(t / G), g = (int)(t % G);
  int s, d;
  if (e < E) { s = ei[e]; d = ei[E + e]; } else { s = d = e - E; }
  float norm = dinv[s] * dinv[d];
  const float4 v = *(const float4*)(hT + (size_t)s * C + g * 4);
  float* o = out + (size_t)d * C + g * 4;
  atomicAdd(o + 0, v.x * norm);
  atomicAdd(o + 1, v.y * norm);
  atomicAdd(o + 2, v.z * norm);
  atomicAdd(o + 3, v.w * norm);
}

// ---------------- bias (+ optional relu), in place ----------------
__global__ void k_bias_act(float* __restrict__ h, const float* __restrict__ b,
                           int C, long long total, int relu) {
  long long t = (long long)blockIdx.x * blockDim.x + threadIdx.x;
  if (t >= total) return;
  float v = h[t] + b[(int)(t % C)];
  h[t] = relu ? fmaxf(v, 0.f) : v;
}

// ---------------- GAT attention logits per (node, head) ----------------
template <int H, int C>
__global__ void k_att_logits(const float* __restrict__ hT,
                             const float* __restrict__ a_s, const float* __restrict__ a_d,
                             float* __restrict__ als, float* __restrict__ ald, int n) {
  int t = blockIdx.x * blockDim.x + threadIdx.x;
  if (t >= n * H) return;
  int node = t / H, hd = t % H;
  const float* hv = hT + (size_t)node * H * C + (size_t)hd * C;
  float ss = 0.f, sd = 0.f;
#pragma unroll
  for (int c = 0; c < C; ++c) {
    float x = hv[c];
    ss += x * a_s[hd * C + c];
    sd += x * a_d[hd * C + c];
  }
  als[t] = ss; ald[t] = sd;
}

__device__ __forceinline__ float leaky02(float x) { return x > 0.f ? x : 0.2f * x; }

__device__ __forceinline__ void atomicMaxFloat(float* addr, float val) {
  // sign-aware int trick: positives via signed max, negatives via unsigned min
  if (val >= 0.f) atomicMax((int*)addr, __float_as_int(val));
  else            atomicMin((unsigned int*)addr, __float_as_uint(val));
}

template <int H>
__global__ void k_gat_max(const int* __restrict__ ei, const float* __restrict__ als,
                          const float* __restrict__ ald, float* __restrict__ mmax,
                          int E, int n) {
  long long t = (long long)blockIdx.x * blockDim.x + threadIdx.x;
  long long total = (long long)(E + n) * H;
  if (t >= total) return;
  int e = (int)(t / H), hd = (int)(t % H);
  int s, d;
  if (e < E) { s = ei[e]; d = ei[E + e]; } else { s = d = e - E; }
  atomicMaxFloat(&mmax[d * H + hd], leaky02(als[s * H + hd] + ald[d * H + hd]));
}

template <int H>
__global__ void k_gat_sum(const int* __restrict__ ei, const float* __restrict__ als,
                          const float* __restrict__ ald, const float* __restrict__ mmax,
                          float* __restrict__ denom, int E, int n) {
  long long t = (long long)blockIdx.x * blockDim.x + threadIdx.x;
  long long total = (long long)(E + n) * H;
  if (t >= total) return;
  int e = (int)(t / H), hd = (int)(t % H);
  int s, d;
  if (e < E) { s = ei[e]; d = ei[E + e]; } else { s = d = e - E; }
  float x = leaky02(als[s * H + hd] + ald[d * H + hd]);
  atomicAdd(&denom[d * H + hd], __expf(x - mmax[d * H + hd]));
}

template <int H, int C>
__global__ void k_gat_msg(const int* __restrict__ ei, const float* __restrict__ als,
                          const float* __restrict__ ald, const float* __restrict__ mmax,
                          const float* __restrict__ denom, const float* __restrict__ hT,
                          float* __restrict__ out, int E, int n) {
  constexpr int G = C / 4;
  long long t = (long long)blockIdx.x * blockDim.x + threadIdx.x;
  long long total = (long long)(E + n) * H * G;
  if (t >= total) return;
  int e = (int)(t / (H * G));
  int r = (int)(t % (H * G));
  int hd = r / G, g = r % G;
  int s, d;
  if (e < E) { s = ei[e]; d = ei[E + e]; } else { s = d = e - E; }
  float x = leaky02(als[s * H + hd] + ald[d * H + hd]);
  float alpha = __expf(x - mmax[d * H + hd]) / (denom[d * H + hd] + 1e-16f);
  const float4 v = *(const float4*)(hT + (size_t)s * H * C + (size_t)hd * C + g * 4);
  float* o = out + (size_t)d * H * C + (size_t)hd * C + g * 4;
  atomicAdd(o + 0, v.x * alpha);
  atomicAdd(o + 1, v.y * alpha);
  atomicAdd(o + 2, v.z * alpha);
  atomicAdd(o + 3, v.w * alpha);
}

// ---------------- prediction head ----------------
__global__ void k_head(const float* __restrict__ emb, const int* __restrict__ ia,
                       const int* __restrict__ ib, const float* __restrict__ W1,
                       const float* __restrict__ b1, const float* __restrict__ W2,
                       const float* __restrict__ b2, float* __restrict__ out, int B) {
  int i = blockIdx.x * blockDim.x + threadIdx.x;
  if (i >= B) return;
  const float* ea = emb + (size_t)ia[i] * 16;
  const float* eb = emb + (size_t)ib[i] * 16;
  float logit = b2[0];
#pragma unroll
  for (int j = 0; j < 16; ++j) {
    float s = b1[j];
#pragma unroll
    for (int k = 0; k < 16; ++k) s += ea[k] * W1[k * 16 + j];
#pragma unroll
    for (int k = 0; k < 16; ++k) s += eb[k] * W1[(16 + k) * 16 + j];
    logit += fmaxf(s, 0.f) * W2[j];
  }
  out[i] = 1.0f / (1.0f + __expf(-logit));
}

// ---------------------------------------------------------------------------
static inline int nblk(long long total, int bs) { return (int)((total + bs - 1) / bs); }

extern "C" void kernel_launch(void* const* d_in, const int* in_sizes, int n_in,
                              void* d_out, int out_size, void* d_ws, size_t ws_size,
                              hipStream_t stream) {
  const float* x        = (const float*)d_in[0];
  const int*   ei       = (const int*)d_in[1];
  const int*   team_a   = (const int*)d_in[2];
  const int*   team_b   = (const int*)d_in[3];
  const float* gcn1_W   = (const float*)d_in[4];
  const float* gcn1_b   = (const float*)d_in[5];
  const float* gcn2_W   = (const float*)d_in[6];
  const float* gcn2_b   = (const float*)d_in[7];
  const float* gat1_W   = (const float*)d_in[8];
  const float* gat1_as  = (const float*)d_in[9];
  const float* gat1_ad  = (const float*)d_in[10];
  const float* gat1_b   = (const float*)d_in[11];
  const float* gat2_W   = (const float*)d_in[12];
  const float* gat2_as  = (const float*)d_in[13];
  const float* gat2_ad  = (const float*)d_in[14];
  const float* gat2_b   = (const float*)d_in[15];
  const float* head_W1  = (const float*)d_in[16];
  const float* head_b1  = (const float*)d_in[17];
  const float* head_W2  = (const float*)d_in[18];
  const float* head_b2  = (const float*)d_in[19];
  float* out = (float*)d_out;

  const int N = in_sizes[0] / 10;     // 50000
  const int E = in_sizes[1] / 2;      // 1600000
  const int B = in_sizes[2];          // 4096
  const long long ET = (long long)E + N;

  // ---- workspace carve-out ----
  char* w = (char*)d_ws;
  size_t off = 0;
  auto alloc = [&](size_t bytes) -> float* {
    float* p = (float*)(w + off);
    off += (bytes + 255) & ~(size_t)255;
    return p;
  };
  float* dinv  = alloc((size_t)N * 4);          // degree -> 1/sqrt(deg)
  float* bufT  = alloc((size_t)N * 128 * 4);    // transform output (x@W)
  float* bufA  = alloc((size_t)N * 128 * 4);    // activations (rotating)
  float* bufB  = alloc((size_t)N * 128 * 4);    // activations (rotating)
  float* als   = alloc((size_t)N * 4 * 4);      // attention logits (src side)
  float* ald   = alloc((size_t)N * 4 * 4);      // attention logits (dst side)
  float* mmax  = alloc((size_t)N * 4 * 4);      // segment max
  float* denom = alloc((size_t)N * 4 * 4);      // segment sum
  (void)ws_size;

  const int BS = 256;
  const int gemmBlocks = ((N + 15) / 16 + 3) / 4;   // 4 waves/block, 1 M-tile/wave

  // ---- degrees & symmetric normalization ----
  hipMemsetAsync(dinv, 0, (size_t)N * 4, stream);
  k_deg<<<nblk(E, BS), BS, 0, stream>>>(ei + E, dinv, E);   // dst row of edge_index
  k_dinv<<<nblk(N, BS), BS, 0, stream>>>(dinv, N);

  // ---- GCN layer 1: h1 = relu(agg(x@W1) + b1), C=32 ----
  k_gemm_f32<10, 32><<<gemmBlocks, 128, 0, stream>>>(x, gcn1_W, bufT, N);
  hipMemsetAsync(bufA, 0, (size_t)N * 32 * 4, stream);
  k_gcn_agg<32><<<nblk(ET * 8, BS), BS, 0, stream>>>(ei, dinv, bufT, bufA, E, N);
  k_bias_act<<<nblk((long long)N * 32, BS), BS, 0, stream>>>(bufA, gcn1_b, 32, (long long)N * 32, 1);

  // ---- GCN layer 2: h2 = relu(agg(h1@W2) + b2), C=32 ----
  k_gemm_f32<32, 32><<<gemmBlocks, 128, 0, stream>>>(bufA, gcn2_W, bufT, N);
  hipMemsetAsync(bufB, 0, (size_t)N * 32 * 4, stream);
  k_gcn_agg<32><<<nblk(ET * 8, BS), BS, 0, stream>>>(ei, dinv, bufT, bufB, E, N);
  k_bias_act<<<nblk((long long)N * 32, BS), BS, 0, stream>>>(bufB, gcn2_b, 32, (long long)N * 32, 1);

  // ---- GAT layer 1: 4 heads x 32, concat -> 128, relu ----
  k_gemm_f32<32, 128><<<gemmBlocks, 128, 0, stream>>>(bufB, gat1_W, bufT, N);
  k_att_logits<4, 32><<<nblk((long long)N * 4, BS), BS, 0, stream>>>(bufT, gat1_as, gat1_ad, als, ald, N);
  k_fill<<<nblk((long long)N * 4, BS), BS, 0, stream>>>(mmax, -INFINITY, (long long)N * 4);
  hipMemsetAsync(denom, 0, (size_t)N * 4 * 4, stream);
  k_gat_max<4><<<nblk(ET * 4, BS), BS, 0, stream>>>(ei, als, ald, mmax, E, N);
  k_gat_sum<4><<<nblk(ET * 4, BS), BS, 0, stream>>>(ei, als, ald, mmax, denom, E, N);
  hipMemsetAsync(bufA, 0, (size_t)N * 128 * 4, stream);
  k_gat_msg<4, 32><<<nblk(ET * 4 * 8, BS), BS, 0, stream>>>(ei, als, ald, mmax, denom, bufT, bufA, E, N);
  k_bias_act<<<nblk((long long)N * 128, BS), BS, 0, stream>>>(bufA, gat1_b, 128, (long long)N * 128, 1);

  // ---- GAT layer 2: 1 head x 16, no concat/mean-of-1, bias only ----
  k_gemm_f32<128, 16><<<gemmBlocks, 128, 0, stream>>>(bufA, gat2_W, bufT, N);
  k_att_logits<1, 16><<<nblk((long long)N, BS), BS, 0, stream>>>(bufT, gat2_as, gat2_ad, als, ald, N);
  k_fill<<<nblk((long long)N, BS), BS, 0, stream>>>(mmax, -INFINITY, (long long)N);
  hipMemsetAsync(denom, 0, (size_t)N * 4, stream);
  k_gat_max<1><<<nblk(ET, BS), BS, 0, stream>>>(ei, als, ald, mmax, E, N);
  k_gat_sum<1><<<nblk(ET, BS), BS, 0, stream>>>(ei, als, ald, mmax, denom, E, N);
  hipMemsetAsync(bufB, 0, (size_t)N * 16 * 4, stream);
  k_gat_msg<1, 16><<<nblk(ET * 4, BS), BS, 0, stream>>>(ei, als, ald, mmax, denom, bufT, bufB, E, N);
  k_bias_act<<<nblk((long long)N * 16, BS), BS, 0, stream>>>(bufB, gat2_b, 16, (long long)N * 16, 0);

  // ---- prediction head ----
  k_head<<<nblk(B, BS), BS, 0, stream>>>(bufB, team_a, team_b, head_W1, head_b1,
                                         head_W2, head_b2, out, B);
}